// DisMaxLossFirstPart_30348238914099
// MI455X (gfx1250) — compile-verified
//
#include <hip/hip_runtime.h>
#include <hip/hip_bf16.h>

typedef __attribute__((ext_vector_type(2))) float v2f;
typedef __attribute__((ext_vector_type(4))) float v4f;
typedef __attribute__((ext_vector_type(8))) float v8f;

#define FDIM    256          // feature dim (K)
#define CCOLS   1000         // classes (N)
#define NTILE_C 63           // ceil(1000/16)
#define KSTEPS  64           // 256 / 4 (K per V_WMMA_F32_16X16X4_F32)
#define WG_ROWS 32           // rows of output strip per workgroup
#define LDS_PAD 4            // row stride 260 ≡ 4 (mod 64 banks) -> conflict-free
#define EPSN    1e-12f

// ---------------------------------------------------------------------------
// Kernel 1: inverse L2 norms of prototypes (wave-per-row, shuffle reduce)
// ---------------------------------------------------------------------------
__global__ void __launch_bounds__(256)
proto_invnorm_kernel(const float* __restrict__ protos,
                     float* __restrict__ invpn, int C)
{
    const int row  = blockIdx.x * 8 + (threadIdx.x >> 5);
    const int lane = threadIdx.x & 31;
    if (row >= C) return;

    const v4f* p4 = (const v4f*)(protos + (size_t)row * FDIM);
    float s = 0.f;
    #pragma unroll
    for (int i = 0; i < 2; ++i) {            // 64 float4 per row / 32 lanes
        v4f v = p4[lane + 32 * i];
        s += v.x * v.x + v.y * v.y + v.z * v.z + v.w * v.w;
    }
    #pragma unroll
    for (int off = 16; off > 0; off >>= 1)
        s += __shfl_xor(s, off, 32);
    if (lane == 0)
        invpn[row] = 1.0f / fmaxf(sqrtf(s), EPSN);
}

// ---------------------------------------------------------------------------
// Kernel 2: fused GEMM (WMMA f32 16x16x4, RAW accumulate) + epilogue
//           normalize + iso + row-mean + single-pass write.
// Bilinearity: fn.pn = (f.p) * invnA[m] * invpn[n]  -> scales applied ONLY in
// the epilogue; the k-loop is pure loads + WMMA.
// One workgroup (8 waves) owns a 32-row strip across all 1000 columns.
//   wave -> row strip (wave>>2)*16, column tiles t = (wave&3) + 4*i, i<16
//   iso kept in VGPRs (16 tiles * v8f); row sums via LDS ds_add_f32 atomics.
// ---------------------------------------------------------------------------
__global__ void __launch_bounds__(256)
dismax_gemm_kernel(const float* __restrict__ feats,
                   const float* __restrict__ protos,
                   const float* __restrict__ invpn,
                   const float* __restrict__ dscale,
                   const float* __restrict__ temp,
                   float* __restrict__ out)
{
    __shared__ float As[WG_ROWS][FDIM + LDS_PAD];   // raw A strip
    __shared__ float norm2[WG_ROWS];
    __shared__ float invnA[WG_ROWS];
    __shared__ float rowsum[WG_ROWS];

    const int tid     = threadIdx.x;
    const int wave    = tid >> 5;
    const int lane    = tid & 31;
    const int rowbase = blockIdx.x * WG_ROWS;

    // ---- stage A strip: 32 rows x 256 f32 = 2048 float4, 8 per thread ----
    {
        const v4f* src = (const v4f*)(feats + (size_t)rowbase * FDIM);
        #pragma unroll
        for (int i = 0; i < 8; ++i) {
            const int idx = tid + 256 * i;   // float4 index in strip
            const int r   = idx >> 6;        // 64 float4 per row
            const int c   = idx & 63;
            v4f v = src[idx];
            *(v4f*)&As[r][c * 4] = v;        // 16B aligned (stride 1040B)
        }
    }
    if (tid < WG_ROWS) { norm2[tid] = 0.f; rowsum[tid] = 0.f; }
    __syncthreads();

    // ---- feature row norms: 8 threads per row, 32 elems each ----
    {
        const int r  = tid >> 3;
        const int c0 = (tid & 7) * 32;
        float s = 0.f;
        #pragma unroll
        for (int j = 0; j < 32; ++j) { float v = As[r][c0 + j]; s += v * v; }
        atomicAdd(&norm2[r], s);             // ds_add_f32
    }
    __syncthreads();
    if (tid < WG_ROWS)
        invnA[tid] = 1.0f / fmaxf(sqrtf(norm2[tid]), EPSN);
    __syncthreads();

    const float absScale = fabsf(dscale[0]);
    const float invTemp  = 1.0f / temp[0];

    // Fragment geometry (ISA 7.12.2, wave32):
    //  A 16x4 f32 : lane holds row M=lane&15; VGPR0/1 = K {0,1} (lanes<16) or {2,3}
    //  B  4x16 f32: lane holds col N=lane&15; VGPR0/1 = K {0,1} (lanes<16) or {2,3}
    //  C 16x16 f32: VGPR r -> row r (lanes<16) / row 8+r (lanes>=16), col lane&15
    const int strip  = wave >> 2;                 // 0..1 -> 16-row sub-strip
    const int tphase = wave & 3;                  // column-tile phase
    const int mrow   = strip * 16 + (lane & 15);  // A row within As
    const int khalf  = (lane < 16) ? 0 : 2;
    const int mbase  = strip * 16 + ((lane < 16) ? 0 : 8);

    // hoist per-lane row scales (constant across all column tiles)
    float invA_r[8];
    #pragma unroll
    for (int r = 0; r < 8; ++r) invA_r[r] = invnA[mbase + r];

    v8f iso[16];

    #pragma unroll
    for (int i = 0; i < 16; ++i) {
        const int t = tphase + 4 * i;             // wave-uniform -> no divergence
        if (t < NTILE_C) {
            const int  col0  = t * 16;
            const int  n     = col0 + (lane & 15);
            const bool valid = (n < CCOLS);
            const int  nc    = valid ? n : (CCOLS - 1);
            const float sB   = invpn[nc];
            const float* prow = protos + (size_t)nc * FDIM;
            __builtin_prefetch(prow, 0, 1);       // global_prefetch_b8

            // ---- pure load + WMMA accumulation loop (no VALU scaling) ----
            v8f acc = {};
            #pragma unroll 4
            for (int k = 0; k < KSTEPS; ++k) {
                const int kb = 4 * k + khalf;
                v2f a  = *(const v2f*)&As[mrow][kb];  // ds_load_b64, conflict-free
                v2f bv = *(const v2f*)(prow + kb);    // global_load_b64 (L2-hot)
                acc = __builtin_amdgcn_wmma_f32_16x16x4_f32(
                        false, a, false, bv, (short)0, acc, false, false);
            }

            // ---- epilogue: cos = acc*invnA[m]*invpn[n]; iso = |s|*sqrt(1-cos) ----
            v8f tiso;
            #pragma unroll
            for (int r = 0; r < 8; ++r) {
                const float cosv = acc[r] * invA_r[r] * sB;
                tiso[r] = absScale *
                          __builtin_amdgcn_sqrtf(fmaxf(1.0f - cosv, 0.0f));
            }
            iso[i] = tiso;

            if (valid) {
                #pragma unroll
                for (int r = 0; r < 8; ++r)
                    atomicAdd(&rowsum[mbase + r], tiso[r]);   // ds_add_f32
            }
        }
    }
    __syncthreads();

    // ---- finalize from registers: out = -(iso + mean)/T ----
    #pragma unroll
    for (int i = 0; i < 16; ++i) {
        const int t = tphase + 4 * i;
        if (t < NTILE_C) {
            const int n = t * 16 + (lane & 15);
            if (n < CCOLS) {
                #pragma unroll
                for (int r = 0; r < 8; ++r) {
                    const int   m    = mbase + r;
                    const float mean = rowsum[m] * (1.0f / CCOLS);
                    out[(size_t)(rowbase + m) * CCOLS + n] =
                        -(iso[i][r] + mean) * invTemp;
                }
            }
        }
    }
}

// ---------------------------------------------------------------------------
extern "C" void kernel_launch(void* const* d_in, const int* in_sizes, int n_in,
                              void* d_out, int out_size, void* d_ws, size_t ws_size,
                              hipStream_t stream)
{
    const float* feats  = (const float*)d_in[0];   // [B, 256] f32
    const float* protos = (const float*)d_in[1];   // [1000, 256] f32
    const float* dscale = (const float*)d_in[2];   // [1] f32
    const float* temp   = (const float*)d_in[3];   // [1] f32
    float*       out    = (float*)d_out;           // [B, 1000] f32
    float*       invpn  = (float*)d_ws;            // [1000] f32 scratch

    const int B = in_sizes[0] / FDIM;              // 32768
    const int C = in_sizes[1] / FDIM;              // 1000

    proto_invnorm_kernel<<<(C + 7) / 8, 256, 0, stream>>>(protos, invpn, C);
    dismax_gemm_kernel<<<B / WG_ROWS, 256, 0, stream>>>(
        feats, protos, invpn, dscale, temp, out);
}